// Hausdorff_Loss_80676665688338
// MI455X (gfx1250) — compile-verified
//
#include <hip/hip_runtime.h>
#include <hip/hip_bf16.h>
#include <stdint.h>

// Hausdorff erosion loss, fused for MI455X (gfx1250).
// B=4, C=2 (use channel 1), D=96, H=192, W=192, K=10 steps, alpha=0.2.
//
// Roofline: ~1.4 GFLOP vs ~1.3 GB moved -> pure bandwidth problem.
// The 2x56.6MB ping-pong working set fits the 192MB L2, so iterations 1..9
// run out of L2; one-shot inputs are streamed non-temporally to protect L2.
// CDNA5 paths used: GLOBAL_LOAD_ASYNC_TO_LDS_B32 (depth-1 software pipeline,
// s_wait_asynccnt), wave32 shuffle reductions, NT cache hints.
//
// d_ws layout:
//   f0   : N floats (ping)   N = 4*96*192*192 = 14,155,776
//   f1   : N floats (pong)
//   mval : K*B floats  (per-iter per-batch max; int-bits atomicMax, deterministic)
//   spart: K*NBLK floats (per-iter per-block partial sums; fixed-order reduce)

#define Bn 4
#define Dn 96
#define Hn 192
#define Wn 192
#define HWn (Hn*Wn)
#define DHWn (Dn*HWn)
#define NTOT (Bn*DHWn)
#define KSTEPS 10
#define CH 48                 // y-rows per block
#define NYCH (Hn/CH)          // 4
#define NBLK (Bn*Dn*NYCH)     // 1536

typedef float v4f __attribute__((ext_vector_type(4)));
typedef int   v4i __attribute__((ext_vector_type(4)));

#define ASYNC_WAIT0() asm volatile("s_wait_asynccnt 0x0" ::: "memory")
#define ASYNC_WAIT1() asm volatile("s_wait_asynccnt 0x1" ::: "memory")

// Issue one async element copy global -> LDS (CDNA5 GLOBAL_LOAD_ASYNC_TO_LDS_B32).
// ldst is a generic pointer into __shared__; its low 32 bits are the group-segment offset.
__device__ __forceinline__ void issue_async_row(const float* __restrict__ src, float* ldst) {
  unsigned lofs = (unsigned)(uintptr_t)(void*)ldst;
  unsigned long long ga = (unsigned long long)(uintptr_t)src;
  asm volatile("global_load_async_to_lds_b32 %0, %1, off"
               :: "v"(lofs), "v"(ga) : "memory");
}

// Pass 0: bound = (output[:,1] - (target==0))^2, float4 vectorized, NT input streams;
// also zeroes the K*B max slots.
__global__ __launch_bounds__(256) void bound_kernel(const float* __restrict__ output,
                                                    const int* __restrict__ target,
                                                    float* __restrict__ f0,
                                                    int* __restrict__ mbits) {
  int i4 = blockIdx.x * 256 + threadIdx.x;        // 0 .. NTOT/4-1 (exact grid)
  int idx = i4 * 4;
  int b = idx / DHWn;
  int r = idx - b * DHWn;
  const v4f p = __builtin_nontemporal_load((const v4f*)(output + (size_t)(b * 2 + 1) * DHWn + r));
  const v4i t = __builtin_nontemporal_load((const v4i*)(target + (size_t)b * DHWn + r));
  v4f o;
  float d;
  d = p.x - ((t.x == 0) ? 1.0f : 0.0f); o.x = d * d;
  d = p.y - ((t.y == 0) ? 1.0f : 0.0f); o.y = d * d;
  d = p.z - ((t.z == 0) ? 1.0f : 0.0f); o.z = d * d;
  d = p.w - ((t.w == 0) ? 1.0f : 0.0f); o.w = d * d;
  *(v4f*)(f0 + idx) = o;                          // RT store: re-read by step 0
  if (i4 < KSTEPS * Bn) mbits[i4] = 0;            // init max slots (0 bits == 0.0f)
}

// One erosion step: out = relu(conv7(in)*cscale - 0.5); unnormalized field kept,
// cscale = (1/m_prev)/7 folds the previous normalization in (conv linearity).
// Rolling 4-slot LDS row window, async global->LDS, one row in flight (pipelined).
template <bool LAST>
__global__ __launch_bounds__(192) void stencil_kernel(const float* __restrict__ in,
                                                      float* __restrict__ out,
                                                      const float* __restrict__ m_prev,
                                                      int* __restrict__ m_cur,
                                                      float* __restrict__ s_part) {
  __shared__ float rows[4][Wn + 2];
  __shared__ float wred[12];                      // 6 wave maxes + 6 wave sums
  const int tid = threadIdx.x;                    // 0..191 (x)
  const int bid = blockIdx.x;
  const int ych = bid % NYCH;
  const int z   = (bid / NYCH) % Dn;
  const int b   = bid / (NYCH * Dn);
  const int y0  = ych * CH;
  const float* plane  = in  + (size_t)b * DHWn + (size_t)z * HWn;
  float*       oplane = out + (size_t)b * DHWn + (size_t)z * HWn;

  float inv_m = 1.0f;
  if (m_prev) { float m = m_prev[b]; inv_m = (m == 0.0f) ? 1.0f : 1.0f / m; }
  const float cscale = inv_m * (1.0f / 7.0f);

  if (tid == 0) {                                 // x halo padding, written once
    #pragma unroll
    for (int s = 0; s < 4; ++s) { rows[s][0] = 0.0f; rows[s][Wn + 1] = 0.0f; }
  }
  // Row r lives in slot (r+1)&3. Preload rows y0-1, y0, y0+1; drain, then barrier.
  #pragma unroll
  for (int r = -1; r <= 1; ++r) {
    const int y = y0 + r;
    float* ld = &rows[(y + 1) & 3][1 + tid];
    if ((unsigned)y < (unsigned)Hn) issue_async_row(plane + (size_t)y * Wn + tid, ld);
    else                            *ld = 0.0f;
  }
  ASYNC_WAIT0();
  __syncthreads();

  float lmax = 0.0f, lsum = 0.0f;
  #pragma unroll 4
  for (int y = y0; y < y0 + CH; ++y) {
    // Pipeline: issue row y+2 (slot previously held row y-2; its readers finished
    // at the end-of-step-(y-1) barrier), then wait so that row y+1 has landed
    // while row y+2 may remain in flight.
    const int yn = y + 2;
    if (yn <= y0 + CH) {
      float* ld = &rows[(yn + 1) & 3][1 + tid];
      if ((unsigned)yn < (unsigned)Hn) {
        issue_async_row(plane + (size_t)yn * Wn + tid, ld);
        ASYNC_WAIT1();                            // row y+1 done, y+2 in flight
      } else {
        *ld = 0.0f;
        ASYNC_WAIT0();                            // nothing newer async: drain y+1
      }
    } else {
      ASYNC_WAIT0();                              // last step: drain row y+1
    }
    __syncthreads();

    const float c  = rows[(y + 1) & 3][tid + 1];
    const float xm = rows[(y + 1) & 3][tid];
    const float xp = rows[(y + 1) & 3][tid + 2];
    const float ym = rows[(y    ) & 3][tid + 1];
    const float yp = rows[(y + 2) & 3][tid + 1];
    const int ro = y * Wn + tid;
    const float zm = (z > 0)      ? plane[ro - HWn] : 0.0f;
    const float zp = (z < Dn - 1) ? plane[ro + HWn] : 0.0f;
    const float s7 = ((c + xm) + (xp + ym)) + ((yp + zm) + zp);
    const float un = fmaxf(fmaf(s7, cscale, -0.5f), 0.0f);
    if (LAST) __builtin_nontemporal_store(un, oplane + ro);   // dead after finalize
    else      oplane[ro] = un;                                // re-read next step (L2)
    lmax = fmaxf(lmax, un);
    lsum += un;
    __syncthreads();                              // readers done before slot reuse
  }

  // wave32 shuffle reduction, then cross-wave via LDS
  for (int off = 16; off > 0; off >>= 1) {
    lmax = fmaxf(lmax, __shfl_down(lmax, off));
    lsum += __shfl_down(lsum, off);
  }
  const int wave = tid >> 5, lane = tid & 31;
  if (lane == 0) { wred[wave] = lmax; wred[6 + wave] = lsum; }
  __syncthreads();
  if (tid == 0) {
    float M = wred[0], S = wred[6];
    #pragma unroll
    for (int w = 1; w < 6; ++w) { M = fmaxf(M, wred[w]); S += wred[6 + w]; }
    // non-negative floats: int compare == float compare; max is order-independent
    atomicMax(m_cur + b, __float_as_int(M));
    s_part[bid] = S;                              // summed in fixed order by finalize
  }
}

// loss = (1/N) * sum_{k,b} (k+1)^0.2 * S_{k,b} / (m_{k,b}==0 ? 1 : m_{k,b})
__global__ __launch_bounds__(64) void finalize_kernel(const float* __restrict__ mvals,
                                                      const float* __restrict__ spart,
                                                      float* __restrict__ outp) {
  __shared__ float contrib[KSTEPS * Bn];
  const int t = threadIdx.x;
  if (t < KSTEPS * Bn) {
    const int k = t / Bn, b = t % Bn;
    const float* sp = spart + (size_t)k * NBLK + (size_t)b * (Dn * NYCH);
    float S = 0.0f;
    for (int i = 0; i < Dn * NYCH; ++i) S += sp[i];   // fixed order
    const float m  = mvals[k * Bn + b];
    const float md = (m == 0.0f) ? 1.0f : m;
    contrib[t] = powf((float)(k + 1), 0.2f) * (S / md);
  }
  __syncthreads();
  if (t == 0) {
    float tot = 0.0f;
    for (int i = 0; i < KSTEPS * Bn; ++i) tot += contrib[i];  // fixed order
    outp[0] = tot / (float)NTOT;
  }
}

extern "C" void kernel_launch(void* const* d_in, const int* in_sizes, int n_in,
                              void* d_out, int out_size, void* d_ws, size_t ws_size,
                              hipStream_t stream) {
  const float* output = (const float*)d_in[0];
  const int*   target = (const int*)d_in[1];
  float* f0    = (float*)d_ws;
  float* f1    = f0 + (size_t)NTOT;
  float* mvals = f1 + (size_t)NTOT;
  float* spart = mvals + KSTEPS * Bn;

  bound_kernel<<<(NTOT / 4) / 256, 256, 0, stream>>>(output, target, f0, (int*)mvals);

  for (int k = 0; k < KSTEPS; ++k) {
    const float* fin  = (k & 1) ? f1 : f0;
    float*       fout = (k & 1) ? f0 : f1;
    const float* mp   = (k == 0) ? nullptr : (mvals + (k - 1) * Bn);
    int*   mc = (int*)(mvals + k * Bn);
    float* sp = spart + (size_t)k * NBLK;
    if (k == KSTEPS - 1) stencil_kernel<true ><<<NBLK, 192, 0, stream>>>(fin, fout, mp, mc, sp);
    else                 stencil_kernel<false><<<NBLK, 192, 0, stream>>>(fin, fout, mp, mc, sp);
  }

  finalize_kernel<<<1, 64, 0, stream>>>(mvals, spart, (float*)d_out);
}